// Grid_11141145166502
// MI455X (gfx1250) — compile-verified
//
#include <hip/hip_runtime.h>
#include <cstdint>

// InstantNGP hash-grid trilinear interpolation, level res=512, T=2^19, F=2.
// Memory-bound gather kernel: table (4 MB) lives in L2 (192 MB); X/out streams
// use non-temporal hints so they don't evict it. No matrix structure -> no WMMA.

#define HASH_T     524288u
#define HASH_MASK  (HASH_T - 1u)
#define PRIME_Y    2654435761u
#define PRIME_Z    805459861u

__global__ __launch_bounds__(256) void hashgrid_trilerp_kernel(
    const float*  __restrict__ X,     // (N,3) in [-1,1]
    const float2* __restrict__ tbl,   // (T,2)
    float2*       __restrict__ out,   // (N,2)
    int n)
{
    int i = blockIdx.x * blockDim.x + threadIdx.x;
    if (i >= n) return;

    const float* xp = X + 3ll * (long long)i;

    // Run-ahead prefetch of the X stream (emits global_prefetch_b8).
    __builtin_prefetch(xp + 3 * 8192, 0, 0);

    // Streaming read of the position: non-temporal, don't pollute caches.
    float x = __builtin_nontemporal_load(xp + 0);
    float y = __builtin_nontemporal_load(xp + 1);
    float z = __builtin_nontemporal_load(xp + 2);

    // Scale [-1,1] -> [0, 511]. Keep the reference's op order/rounding.
    float xs = (x + 1.0f) * 0.5f * 511.0f;
    float ys = (y + 1.0f) * 0.5f * 511.0f;
    float zs = (z + 1.0f) * 0.5f * 511.0f;

    float fx = floorf(xs), fy = floorf(ys), fz = floorf(zs);
    float w0 = xs - fx,    w1 = ys - fy,    w2 = zs - fz;

    // Reference's ceil-with-fixup is always floor+1. Hash needs only the low
    // 19 bits, so uint32 arithmetic is exact (mod-2^19 == & MASK; coords >= 0).
    uint32_t hx0 = (uint32_t)(int)fx;              // * PRIME_X (=1)
    uint32_t hy0 = (uint32_t)(int)fy * PRIME_Y;
    uint32_t hz0 = (uint32_t)(int)fz * PRIME_Z;
    uint32_t hx1 = hx0 + 1u;
    uint32_t hy1 = hy0 + PRIME_Y;                  // (fy+1)*P mod 2^32
    uint32_t hz1 = hz0 + PRIME_Z;

    // Corner k: bit2 -> x (ceil), bit1 -> y, bit0 -> z.
    uint32_t i0 = (hx0 ^ hy0 ^ hz0) & HASH_MASK;   // k=0: (f,f,f)
    uint32_t i1 = (hx0 ^ hy0 ^ hz1) & HASH_MASK;   // k=1: (f,f,c)
    uint32_t i2 = (hx0 ^ hy1 ^ hz0) & HASH_MASK;   // k=2: (f,c,f)
    uint32_t i3 = (hx0 ^ hy1 ^ hz1) & HASH_MASK;   // k=3: (f,c,c)
    uint32_t i4 = (hx1 ^ hy0 ^ hz0) & HASH_MASK;   // k=4: (c,f,f)
    uint32_t i5 = (hx1 ^ hy0 ^ hz1) & HASH_MASK;   // k=5: (c,f,c)
    uint32_t i6 = (hx1 ^ hy1 ^ hz0) & HASH_MASK;   // k=6: (c,c,f)
    uint32_t i7 = (hx1 ^ hy1 ^ hz1) & HASH_MASK;   // k=7: (c,c,c)

    // 8 independent b64 gathers -> high MLP per wave; default (RT) temporal
    // hint so the 4 MB table stays resident in L2.
    float2 v0 = tbl[i0];
    float2 v1 = tbl[i1];
    float2 v2 = tbl[i2];
    float2 v3 = tbl[i3];
    float2 v4 = tbl[i4];
    float2 v5 = tbl[i5];
    float2 v6 = tbl[i6];
    float2 v7 = tbl[i7];

    // Trilinear interpolation, reference form a*(1-w) + b*w per feature.
    float u0 = 1.0f - w0, u1 = 1.0f - w1, u2 = 1.0f - w2;

    float p0x = v0.x * u0 + v4.x * w0;
    float p0y = v0.y * u0 + v4.y * w0;
    float p1x = v1.x * u0 + v5.x * w0;
    float p1y = v1.y * u0 + v5.y * w0;
    float p2x = v2.x * u0 + v6.x * w0;
    float p2y = v2.y * u0 + v6.y * w0;
    float p3x = v3.x * u0 + v7.x * w0;
    float p3y = v3.y * u0 + v7.y * w0;

    float q0x = p0x * u1 + p2x * w1;
    float q0y = p0y * u1 + p2y * w1;
    float q1x = p1x * u1 + p3x * w1;
    float q1y = p1y * u1 + p3y * w1;

    float ox = q0x * u2 + q1x * w2;
    float oy = q0y * u2 + q1y * w2;

    // Streaming write: non-temporal (b64 NT store), output never re-read.
    float* op = (float*)(out + i);
    __builtin_nontemporal_store(ox, op + 0);
    __builtin_nontemporal_store(oy, op + 1);
}

extern "C" void kernel_launch(void* const* d_in, const int* in_sizes, int n_in,
                              void* d_out, int out_size, void* d_ws, size_t ws_size,
                              hipStream_t stream) {
    const float*  X   = (const float*)d_in[0];    // (N,3) float32
    const float2* tbl = (const float2*)d_in[1];   // (T,2) float32
    float2*       out = (float2*)d_out;           // (N,2) float32

    int n = in_sizes[0] / 3;                      // N = 4,194,304
    int block = 256;                              // 8 wave32 waves per block
    int grid  = (n + block - 1) / block;          // 16384 workgroups

    hashgrid_trilerp_kernel<<<grid, block, 0, stream>>>(X, tbl, out, n);
}